// GraNNy_ViPeR_70325794505175
// MI455X (gfx1250) — compile-verified
//
#include <hip/hip_runtime.h>
#include <hip/hip_bf16.h>

typedef __attribute__((ext_vector_type(2))) float v2f;
typedef __attribute__((ext_vector_type(8))) float v8f;

#define HAVE_WMMA_F32 __has_builtin(__builtin_amdgcn_wmma_f32_16x16x4_f32)

// ---------------------------------------------------------------------------
// degree / normalization
// ---------------------------------------------------------------------------
__global__ void deg_kernel(const int* __restrict__ dst, float* __restrict__ deg, int E) {
    int e = blockIdx.x * blockDim.x + threadIdx.x;
    if (e < E) atomicAdd(&deg[dst[e]], 1.0f);
}

__global__ void invdeg_kernel(const float* __restrict__ deg, float* __restrict__ invd,
                              float* __restrict__ dinv, int N) {
    int i = blockIdx.x * blockDim.x + threadIdx.x;
    if (i >= N) return;
    float d = deg[i];
    invd[i] = d > 0.0f ? 1.0f / d : 0.0f;
    dinv[i] = d > 0.0f ? rsqrtf(d) : 0.0f;
}

// ---------------------------------------------------------------------------
// edge scatter kernels (atomic f32 adds)
// ---------------------------------------------------------------------------
__global__ void scatter8(const int* __restrict__ src, const int* __restrict__ dst,
                         const float* __restrict__ X, float* __restrict__ agg, int E) {
    int e = blockIdx.x * blockDim.x + threadIdx.x;
    if (e >= E) return;
    int s = src[e], d = dst[e];
    const float4* xp = (const float4*)(X + (size_t)s * 8);
    float4 x0 = xp[0], x1 = xp[1];
    float* a = agg + (size_t)d * 8;
    atomicAdd(a + 0, x0.x); atomicAdd(a + 1, x0.y);
    atomicAdd(a + 2, x0.z); atomicAdd(a + 3, x0.w);
    atomicAdd(a + 4, x1.x); atomicAdd(a + 5, x1.y);
    atomicAdd(a + 6, x1.z); atomicAdd(a + 7, x1.w);
}

// one wave (32 lanes) per edge: coalesced 512B gather, 128 lane-parallel atomics
__global__ void scatter128(const int* __restrict__ src, const int* __restrict__ dst,
                           const float* __restrict__ H, float* __restrict__ agg, int E) {
    int e = blockIdx.x * (blockDim.x >> 5) + (threadIdx.x >> 5);
    int lane = threadIdx.x & 31;
    if (e >= E) return;
    int s = src[e], d = dst[e];
    float4 v = ((const float4*)(H + (size_t)s * 128))[lane];
    float* a = agg + (size_t)d * 128 + lane * 4;
    atomicAdd(a + 0, v.x); atomicAdd(a + 1, v.y);
    atomicAdd(a + 2, v.z); atomicAdd(a + 3, v.w);
}

__global__ void scatter1(const int* __restrict__ src, const int* __restrict__ dst,
                         const float* __restrict__ t, float* __restrict__ agg, int E) {
    int e = blockIdx.x * blockDim.x + threadIdx.x;
    if (e >= E) return;
    atomicAdd(&agg[dst[e]], t[src[e]]);
}

// TAG propagation: nxt[dst] += dinv[src]*dinv[dst] * cur[src]  (8 features)
__global__ void tag_prop(const int* __restrict__ src, const int* __restrict__ dst,
                         const float* __restrict__ dinv, const float* __restrict__ cur,
                         float* __restrict__ nxt, int E) {
    int e = blockIdx.x * blockDim.x + threadIdx.x;
    if (e >= E) return;
    int s = src[e], d = dst[e];
    float w = dinv[s] * dinv[d];
    const float4* cp = (const float4*)(cur + (size_t)s * 8);
    float4 c0 = cp[0], c1 = cp[1];
    float* n = nxt + (size_t)d * 8;
    atomicAdd(n + 0, w * c0.x); atomicAdd(n + 1, w * c0.y);
    atomicAdd(n + 2, w * c0.z); atomicAdd(n + 3, w * c0.w);
    atomicAdd(n + 4, w * c1.x); atomicAdd(n + 5, w * c1.y);
    atomicAdd(n + 6, w * c1.z); atomicAdd(n + 7, w * c1.w);
}

// ---------------------------------------------------------------------------
// Fused SAGE layer: Out = relu( (Agg * inv_deg) @ Wl  +  H @ Wr  + b )
// Wl,Wr: [K,128] row-major. 256 threads = 8 waves; block owns rows
// [16*blockIdx, +16); wave w owns cols [16w, 16w+16). f32 WMMA 16x16x4.
// In-place safe (Out may alias Agg): all reads precede __syncthreads().
// ---------------------------------------------------------------------------
template <int K>
__global__ void sage_gemm_wmma(const float* __restrict__ Agg, const float* __restrict__ H,
                               const float* __restrict__ Wl, const float* __restrict__ Wr,
                               const float* __restrict__ bias, const float* __restrict__ invd,
                               float* __restrict__ Out, int N) {
    const int wave = threadIdx.x >> 5;
    const int lane = threadIdx.x & 31;
    const int m = lane & 15;            // A row within tile / B col within tile
    const int koff = (lane >> 4) << 1;  // 0 or 2 (A/B K-pair per half-wave)
    const int row0 = blockIdx.x * 16;
    const int row = row0 + m;
    const int col = wave * 16 + m;

    const float scale = invd[row];

#if HAVE_WMMA_F32
    v8f c = {};
    #pragma unroll 4
    for (int k = 0; k < K; k += 4) {
        v2f a, b;
        a.x = Agg[(size_t)row * K + k + koff] * scale;
        a.y = Agg[(size_t)row * K + k + koff + 1] * scale;
        b.x = Wl[(size_t)(k + koff) * 128 + col];
        b.y = Wl[(size_t)(k + koff + 1) * 128 + col];
        c = __builtin_amdgcn_wmma_f32_16x16x4_f32(false, a, false, b, (short)0, c,
                                                  false, false);
    }
    #pragma unroll 4
    for (int k = 0; k < K; k += 4) {
        v2f a, b;
        a.x = H[(size_t)row * K + k + koff];
        a.y = H[(size_t)row * K + k + koff + 1];
        b.x = Wr[(size_t)(k + koff) * 128 + col];
        b.y = Wr[(size_t)(k + koff + 1) * 128 + col];
        c = __builtin_amdgcn_wmma_f32_16x16x4_f32(false, a, false, b, (short)0, c,
                                                  false, false);
    }
    const float bcol = bias[col];
    __syncthreads();  // in-place safety: all Agg reads complete before any store
    const int half = (lane >> 4) * 8;  // C layout: lanes 16-31 hold M = r+8
    #pragma unroll
    for (int r = 0; r < 8; ++r) {
        float v = c[r] + bcol;
        Out[(size_t)(row0 + half + r) * 128 + col] = v > 0.0f ? v : 0.0f;
    }
#else
    // scalar fallback: each lane computes rows [row0..row0+16) x one column
    float acc[16];
    #pragma unroll
    for (int r = 0; r < 16; ++r) acc[r] = bias[col];
    for (int k = 0; k < K; ++k) {
        float wl = Wl[(size_t)k * 128 + col], wr = Wr[(size_t)k * 128 + col];
        #pragma unroll
        for (int r = 0; r < 16; ++r) {
            size_t ri = (size_t)(row0 + r);
            acc[r] += Agg[ri * K + k] * invd[row0 + r] * wl + H[ri * K + k] * wr;
        }
    }
    __syncthreads();
    #pragma unroll
    for (int r = 0; r < 16; ++r) {
        float v = acc[r];
        Out[(size_t)(row0 + r) * 128 + col] = v > 0.0f ? v : 0.0f;
    }
#endif
}

// t3 = H @ W3l ; r3 = H @ W3r   (H: [N,128], W: [128]) — one wave per node
__global__ void dot128_dual(const float* __restrict__ H, const float* __restrict__ wl,
                            const float* __restrict__ wr, float* __restrict__ t3,
                            float* __restrict__ r3, int N) {
    int node = blockIdx.x * (blockDim.x >> 5) + (threadIdx.x >> 5);
    int lane = threadIdx.x & 31;
    if (node >= N) return;
    float4 h = ((const float4*)(H + (size_t)node * 128))[lane];
    float4 a = ((const float4*)wl)[lane];
    float4 b = ((const float4*)wr)[lane];
    float sl = h.x * a.x + h.y * a.y + h.z * a.z + h.w * a.w;
    float sr = h.x * b.x + h.y * b.y + h.z * b.z + h.w * b.w;
    #pragma unroll
    for (int off = 16; off > 0; off >>= 1) {
        sl += __shfl_down(sl, off, 32);
        sr += __shfl_down(sr, off, 32);
    }
    if (lane == 0) { t3[node] = sl; r3[node] = sr; }
}

// acc[i,:8] += h[i,:8] @ W  (W: [8,8] row-major)
__global__ void tag_acc8(const float* __restrict__ h, const float* __restrict__ W,
                         float* __restrict__ acc, int N) {
    int i = blockIdx.x * blockDim.x + threadIdx.x;
    if (i >= N) return;
    const float4* hp = (const float4*)(h + (size_t)i * 8);
    float4 h0 = hp[0], h1 = hp[1];
    float hv[8] = {h0.x, h0.y, h0.z, h0.w, h1.x, h1.y, h1.z, h1.w};
    float4* ap = (float4*)(acc + (size_t)i * 8);
    float4 a0 = ap[0], a1 = ap[1];
    float o[8] = {a0.x, a0.y, a0.z, a0.w, a1.x, a1.y, a1.z, a1.w};
    #pragma unroll
    for (int f = 0; f < 8; ++f) {
        float x = hv[f];
        #pragma unroll
        for (int oc = 0; oc < 8; ++oc) o[oc] += x * W[f * 8 + oc];
    }
    ap[0] = make_float4(o[0], o[1], o[2], o[3]);
    ap[1] = make_float4(o[4], o[5], o[6], o[7]);
}

// acc[i] += dot(h[i,:8], W[:8])
__global__ void tag_acc1(const float* __restrict__ h, const float* __restrict__ W,
                         float* __restrict__ acc, int N) {
    int i = blockIdx.x * blockDim.x + threadIdx.x;
    if (i >= N) return;
    const float4* hp = (const float4*)(h + (size_t)i * 8);
    float4 h0 = hp[0], h1 = hp[1];
    float s = h0.x * W[0] + h0.y * W[1] + h0.z * W[2] + h0.w * W[3] +
              h1.x * W[4] + h1.y * W[5] + h1.z * W[6] + h1.w * W[7];
    acc[i] += s;
}

__global__ void relu_bias8(float* __restrict__ x, const float* __restrict__ b, int N) {
    int t = blockIdx.x * blockDim.x + threadIdx.x;
    if (t >= N * 8) return;
    int f = t & 7;
    float v = x[t] + b[f];
    x[t] = v > 0.0f ? v : 0.0f;
}

// final fusion: SAGE-3 epilogue + TAG biases + lin2 + lin1
__global__ void final_k(const float* __restrict__ aggS, const float* __restrict__ invd,
                        const float* __restrict__ r3, const float* __restrict__ b3,
                        const float* __restrict__ a2s, const float* __restrict__ ta2b,
                        const float* __restrict__ a3s, const float* __restrict__ tb2b,
                        const float* __restrict__ l2W, const float* __restrict__ l2b,
                        const float* __restrict__ l1W, const float* __restrict__ l1b,
                        float* __restrict__ out, int N) {
    int i = blockIdx.x * blockDim.x + threadIdx.x;
    if (i >= N) return;
    float x1 = aggS[i] * invd[i] + r3[i] + b3[0];
    x1 = x1 > 0.0f ? x1 : 0.0f;
    float x2 = a2s[i] + ta2b[0]; x2 = x2 > 0.0f ? x2 : 0.0f;
    float x3 = a3s[i] + tb2b[0]; x3 = x3 > 0.0f ? x3 : 0.0f;
    float x23 = x2 * l2W[0] + x3 * l2W[1] + l2b[0];
    x23 = x23 > 0.0f ? x23 : 0.0f;
    float o = x1 * l1W[0] + x23 * l1W[1] + l1b[0];
    out[i] = o > 0.0f ? o : 0.0f;
}

// ---------------------------------------------------------------------------
extern "C" void kernel_launch(void* const* d_in, const int* in_sizes, int n_in,
                              void* d_out, int out_size, void* d_ws, size_t ws_size,
                              hipStream_t stream) {
    const float* x    = (const float*)d_in[0];
    const int*   ei   = (const int*)d_in[1];
    const float* W1l  = (const float*)d_in[2];
    const float* W1r  = (const float*)d_in[3];
    const float* b1   = (const float*)d_in[4];
    const float* W2l  = (const float*)d_in[5];
    const float* W2r  = (const float*)d_in[6];
    const float* b2   = (const float*)d_in[7];
    const float* W3l  = (const float*)d_in[8];
    const float* W3r  = (const float*)d_in[9];
    const float* b3   = (const float*)d_in[10];
    const float* Ta1W = (const float*)d_in[11];
    const float* Ta1b = (const float*)d_in[12];
    const float* Ta2W = (const float*)d_in[13];
    const float* Ta2b = (const float*)d_in[14];
    const float* Tb1W = (const float*)d_in[15];
    const float* Tb1b = (const float*)d_in[16];
    const float* Tb2W = (const float*)d_in[17];
    const float* Tb2b = (const float*)d_in[18];
    const float* l2W  = (const float*)d_in[19];
    const float* l2b  = (const float*)d_in[20];
    const float* l1W  = (const float*)d_in[21];
    const float* l1b  = (const float*)d_in[22];

    const int N = in_sizes[0] / 8;
    const int E = in_sizes[1] / 2;
    const int* src = ei;
    const int* dst = ei + E;

    // workspace layout (floats)
    float* ws   = (float*)d_ws;
    size_t off  = 0;
    float* deg  = ws + off; off += (size_t)N;
    float* invd = ws + off; off += (size_t)N;
    float* dinv = ws + off; off += (size_t)N;
    float* agg8 = ws + off; off += (size_t)N * 8;
    float* h1   = ws + off; off += (size_t)N * 128;
    float* agg1 = ws + off; off += (size_t)N * 128;  // becomes h2 in place
    float* t3   = ws + off; off += (size_t)N;
    float* r3   = ws + off; off += (size_t)N;
    float* aggS = ws + off; off += (size_t)N;
    float* P0   = ws + off; off += (size_t)N * 8;
    float* P1   = ws + off; off += (size_t)N * 8;
    float* acc2 = ws + off; off += (size_t)N * 8;
    float* acc3 = ws + off; off += (size_t)N * 8;
    float* a2s  = ws + off; off += (size_t)N;
    float* a3s  = ws + off; off += (size_t)N;

    const int nb  = (N + 255) / 256;          // node-parallel
    const int n8b = (N * 8 + 255) / 256;      // node*feature parallel
    const int eb  = (E + 255) / 256;          // edge-parallel
    const int ewb = (E + 7) / 8;              // wave-per-edge (256 thr = 8 waves)
    const int nwb = (N + 7) / 8;              // wave-per-node
    const int gb  = N / 16;                   // GEMM row tiles (trailing N % 16 rows not computed)
    const size_t N8bytes = (size_t)N * 8 * sizeof(float);

    // --- degrees & norms ---
    hipMemsetAsync(deg, 0, (size_t)N * sizeof(float), stream);
    deg_kernel<<<eb, 256, 0, stream>>>(dst, deg, E);
    invdeg_kernel<<<nb, 256, 0, stream>>>(deg, invd, dinv, N);

    // --- SAGE layer 1: agg8 = sum x[src]; h1 = relu(agg8*invd @W1l + x@W1r + b1)
    hipMemsetAsync(agg8, 0, N8bytes, stream);
    scatter8<<<eb, 256, 0, stream>>>(src, dst, x, agg8, E);
    sage_gemm_wmma<8><<<gb, 256, 0, stream>>>(agg8, x, W1l, W1r, b1, invd, h1, N);

    // --- SAGE layer 2: agg1 = sum h1[src]; h2 = relu(... ) written over agg1
    hipMemsetAsync(agg1, 0, (size_t)N * 128 * sizeof(float), stream);
    scatter128<<<ewb, 256, 0, stream>>>(src, dst, h1, agg1, E);
    sage_gemm_wmma<128><<<gb, 256, 0, stream>>>(agg1, h1, W2l, W2r, b2, invd, agg1, N);
    const float* h2 = agg1;

    // --- SAGE layer 3 (collapsed): t3 = h2@W3l, r3 = h2@W3r; aggS = scatter t3
    dot128_dual<<<nwb, 256, 0, stream>>>(h2, W3l, W3r, t3, r3, N);
    hipMemsetAsync(aggS, 0, (size_t)N * sizeof(float), stream);
    scatter1<<<eb, 256, 0, stream>>>(src, dst, t3, aggS, E);

    // --- TAG: shared propagation chain P^k x for Ta1 (K=3) and Tb1 (K=9)
    hipMemsetAsync(acc2, 0, N8bytes, stream);
    hipMemsetAsync(acc3, 0, N8bytes, stream);
    tag_acc8<<<nb, 256, 0, stream>>>(x, Ta1W, acc2, N);
    tag_acc8<<<nb, 256, 0, stream>>>(x, Tb1W, acc3, N);
    const float* cur = x;
    for (int k = 1; k <= 9; ++k) {
        float* nxt = (k & 1) ? P0 : P1;
        hipMemsetAsync(nxt, 0, N8bytes, stream);
        tag_prop<<<eb, 256, 0, stream>>>(src, dst, dinv, cur, nxt, E);
        if (k <= 3) tag_acc8<<<nb, 256, 0, stream>>>(nxt, Ta1W + (size_t)k * 64, acc2, N);
        tag_acc8<<<nb, 256, 0, stream>>>(nxt, Tb1W + (size_t)k * 64, acc3, N);
        cur = nxt;
    }
    relu_bias8<<<n8b, 256, 0, stream>>>(acc2, Ta1b, N);  // x2 in acc2
    relu_bias8<<<n8b, 256, 0, stream>>>(acc3, Tb1b, N);  // x3 in acc3

    // --- Ta2 on x2 (K=3, 8->1) ---
    hipMemsetAsync(a2s, 0, (size_t)N * sizeof(float), stream);
    tag_acc1<<<nb, 256, 0, stream>>>(acc2, Ta2W, a2s, N);
    cur = acc2;
    for (int k = 1; k <= 3; ++k) {
        float* nxt = (k & 1) ? P0 : P1;
        hipMemsetAsync(nxt, 0, N8bytes, stream);
        tag_prop<<<eb, 256, 0, stream>>>(src, dst, dinv, cur, nxt, E);
        tag_acc1<<<nb, 256, 0, stream>>>(nxt, Ta2W + (size_t)k * 8, a2s, N);
        cur = nxt;
    }

    // --- Tb2 on x3 (K=9, 8->1) ---
    hipMemsetAsync(a3s, 0, (size_t)N * sizeof(float), stream);
    tag_acc1<<<nb, 256, 0, stream>>>(acc3, Tb2W, a3s, N);
    cur = acc3;
    for (int k = 1; k <= 9; ++k) {
        float* nxt = (k & 1) ? P0 : P1;
        hipMemsetAsync(nxt, 0, N8bytes, stream);
        tag_prop<<<eb, 256, 0, stream>>>(src, dst, dinv, cur, nxt, E);
        tag_acc1<<<nb, 256, 0, stream>>>(nxt, Tb2W + (size_t)k * 8, a3s, N);
        cur = nxt;
    }

    // --- final fusion ---
    final_k<<<nb, 256, 0, stream>>>(aggS, invd, r3, b3, a2s, Ta2b, a3s, Tb2b,
                                    l2W, l2b, l1W, l1b, (float*)d_out, N);
}